// GraphProp_68058051772923
// MI455X (gfx1250) — compile-verified
//
#include <hip/hip_runtime.h>
#include <math.h>

#define ROUNDS   2
#define N_NODES  50000
#define N_EDGES  800000
#define ND       128          // NODE_DIM
#define ED       32           // EDGE_DIM
#define AD       128          // ACT_DIM
#define H3       384          // 3 * NODE_DIM
#define CATD     (2*ND + ED)  // 288

typedef __attribute__((ext_vector_type(2))) float v2f;
typedef __attribute__((ext_vector_type(8))) float v8f;

// ---------------------------------------------------------------------------
// Zero the scatter accumulators (S1 | S3 | deg laid out contiguously in ws)
// ---------------------------------------------------------------------------
__global__ void zero_kernel(float* __restrict__ p, int n) {
    int stride = gridDim.x * blockDim.x;
    for (int i = blockIdx.x * blockDim.x + threadIdx.x; i < n; i += stride)
        p[i] = 0.0f;
}

// ---------------------------------------------------------------------------
// Edge scatter phase: S1[d] += hv[s], S3[d] += he[e], deg[d] += 1
// 4 threads per edge, each covering 32 of the 128 node dims (8 of 32 edge dims)
// hv (25.6MB) and S1/S3 (32MB) live in the 192MB L2 -> atomics stay on-chip.
// ---------------------------------------------------------------------------
__global__ __launch_bounds__(256) void edge_scatter_kernel(
    const float* __restrict__ hv, const float* __restrict__ he,
    const int* __restrict__ src, const int* __restrict__ dst,
    float* __restrict__ S1, float* __restrict__ S3, float* __restrict__ degv,
    int n_edges)
{
    int gid  = blockIdx.x * 256 + threadIdx.x;
    int e    = gid >> 2;
    int part = gid & 3;
    if (e >= n_edges) return;

    int s = src[e];
    int d = dst[e];

    const float* hs  = hv + (size_t)s * ND + part * 32;
    float*       s1d = S1 + (size_t)d * ND + part * 32;
#pragma unroll
    for (int i = 0; i < 32; i += 4) {
        float4 v = *(const float4*)(hs + i);
        atomicAdd(s1d + i + 0, v.x);
        atomicAdd(s1d + i + 1, v.y);
        atomicAdd(s1d + i + 2, v.z);
        atomicAdd(s1d + i + 3, v.w);
    }

    const float* hep = he + (size_t)e * ED + part * 8;
    float*       s3d = S3 + (size_t)d * ED + part * 8;
#pragma unroll
    for (int i = 0; i < 8; i += 4) {
        float4 v = *(const float4*)(hep + i);
        atomicAdd(s3d + i + 0, v.x);
        atomicAdd(s3d + i + 1, v.y);
        atomicAdd(s3d + i + 2, v.z);
        atomicAdd(s3d + i + 3, v.w);
    }

    if (part == 0) atomicAdd(degv + d, 1.0f);
}

// ---------------------------------------------------------------------------
// WMMA helper: C(16x16,f32) += A(16x4 from LDS, row-major, leading dim lda)
//                           x  B(4x16 from global weights, row-major, ld ldw)
// A frag (ISA 16x4 f32 layout): lane m=lane&15, VGPR j holds K = k + 2*(lane>>4) + j
// B frag mirrors C/D striping: lane n=lane&15, VGPR j holds B[k + 2*(lane>>4) + j][n]
// ---------------------------------------------------------------------------
__device__ __forceinline__ void mm_acc(
    const float* __restrict__ A, int lda,
    const float* __restrict__ W, int ldw, int colbase,
    int K, int nl, int koff, v8f& c)
{
#pragma unroll 4
    for (int k = 0; k < K; k += 4) {
        v2f a = *(const v2f*)(A + nl * lda + k + koff);   // 8B-aligned LDS load
        v2f b;
        b.x = W[(size_t)(k + koff)     * ldw + colbase + nl];
        b.y = W[(size_t)(k + koff + 1) * ldw + colbase + nl];
        c = __builtin_amdgcn_wmma_f32_16x16x4_f32(
                /*neg_a=*/false, a, /*neg_b=*/false, b,
                /*c_mod=*/(short)0, c, /*reuse_a=*/false, /*reuse_b=*/false);
    }
}

// ---------------------------------------------------------------------------
// Node phase: per 16-node tile (grid exactly 3125 tiles, always full):
//   a  = S1@Ws + (deg*hv)@Wd + S3@We + deg (x) b_msg
//   gi = a@W_ih + b_ih ; gh = hv@W_hh + b_hh ; GRU gates ; hv_out
// 256 threads = 8 waves; wave w owns output columns [16w, 16w+16).
// ---------------------------------------------------------------------------
__global__ __launch_bounds__(256) void node_update_kernel(
    const float* __restrict__ hv_in,
    const float* __restrict__ S1,
    const float* __restrict__ S3,
    const float* __restrict__ degv,
    const float* __restrict__ Wmsg,   // 288 x 128 (Ws|Wd|We stacked on K)
    const float* __restrict__ bmsg,   // 128
    const float* __restrict__ Wih,    // 128 x 384
    const float* __restrict__ Whh,    // 128 x 384
    const float* __restrict__ bih,    // 384
    const float* __restrict__ bhh,    // 384
    float* __restrict__ hv_out)
{
    __shared__ float hv_s[16 * ND];
    __shared__ float s1_s[16 * ND];
    __shared__ float x2_s[16 * ND];   // deg * hv
    __shared__ float s3_s[16 * ED];
    __shared__ float deg_s[16];
    __shared__ float a_s[16 * AD];

    const int tid  = threadIdx.x;
    const int base = blockIdx.x * 16;

    if (tid < 16) deg_s[tid] = degv[base + tid];
    for (int i = tid; i < 16 * ND; i += 256) {
        int r = i >> 7, col = i & 127;
        size_t g = (size_t)(base + r) * ND + col;
        hv_s[i] = hv_in[g];
        s1_s[i] = S1[g];
    }
    for (int i = tid; i < 16 * ED; i += 256) {
        int r = i >> 5, col = i & 31;
        s3_s[i] = S3[(size_t)(base + r) * ED + col];
    }
    __syncthreads();
    for (int i = tid; i < 16 * ND; i += 256)
        x2_s[i] = hv_s[i] * deg_s[i >> 7];
    __syncthreads();

    const int wave = tid >> 5;
    const int lane = tid & 31;
    const int mh   = lane >> 4;     // half-select (rows 0-7 vs 8-15 per C reg)
    const int nl   = lane & 15;     // column within tile / row of A
    const int koff = mh << 1;       // K sub-offset for A/B fragments
    const int cb   = wave << 4;     // this wave's 16-column slice

    // ---- Phase 2: a = S1@Ws + x2@Wd + S3@We + deg*b_msg --------------------
    {
        v8f c;
        float bm = bmsg[cb + nl];
#pragma unroll
        for (int i = 0; i < 8; ++i) c[i] = deg_s[i + (mh << 3)] * bm;
        mm_acc(s1_s, ND, Wmsg,              AD, cb, ND, nl, koff, c);
        mm_acc(x2_s, ND, Wmsg + ND * AD,    AD, cb, ND, nl, koff, c);
        mm_acc(s3_s, ED, Wmsg + 2 * ND * AD, AD, cb, ED, nl, koff, c);
#pragma unroll
        for (int i = 0; i < 8; ++i)
            a_s[(i + (mh << 3)) * AD + cb + nl] = c[i];
    }
    __syncthreads();

    // ---- Phase 3: GRU for hidden dims [cb, cb+16) --------------------------
    float rg[8], zg[8], ng[8];
    {   // r = sigmoid(ir + hr)
        v8f ci, ch;
        float bi = bih[cb + nl], bh = bhh[cb + nl];
#pragma unroll
        for (int i = 0; i < 8; ++i) { ci[i] = bi; ch[i] = bh; }
        mm_acc(a_s,  AD, Wih, H3, cb, AD, nl, koff, ci);
        mm_acc(hv_s, ND, Whh, H3, cb, ND, nl, koff, ch);
#pragma unroll
        for (int i = 0; i < 8; ++i)
            rg[i] = 1.0f / (1.0f + __expf(-(ci[i] + ch[i])));
    }
    {   // z = sigmoid(iz + hz)
        v8f ci, ch;
        float bi = bih[ND + cb + nl], bh = bhh[ND + cb + nl];
#pragma unroll
        for (int i = 0; i < 8; ++i) { ci[i] = bi; ch[i] = bh; }
        mm_acc(a_s,  AD, Wih, H3, ND + cb, AD, nl, koff, ci);
        mm_acc(hv_s, ND, Whh, H3, ND + cb, ND, nl, koff, ch);
#pragma unroll
        for (int i = 0; i < 8; ++i)
            zg[i] = 1.0f / (1.0f + __expf(-(ci[i] + ch[i])));
    }
    {   // n = tanh(in + r * hn)
        v8f ci, ch;
        float bi = bih[2 * ND + cb + nl], bh = bhh[2 * ND + cb + nl];
#pragma unroll
        for (int i = 0; i < 8; ++i) { ci[i] = bi; ch[i] = bh; }
        mm_acc(a_s,  AD, Wih, H3, 2 * ND + cb, AD, nl, koff, ci);
        mm_acc(hv_s, ND, Whh, H3, 2 * ND + cb, ND, nl, koff, ch);
#pragma unroll
        for (int i = 0; i < 8; ++i)
            ng[i] = tanhf(ci[i] + rg[i] * ch[i]);
    }
    // h' = (1-z)*n + z*h
#pragma unroll
    for (int i = 0; i < 8; ++i) {
        int m = i + (mh << 3);
        float h = hv_s[m * ND + cb + nl];
        hv_out[(size_t)(base + m) * ND + cb + nl] =
            (1.0f - zg[i]) * ng[i] + zg[i] * h;
    }
}

// ---------------------------------------------------------------------------
extern "C" void kernel_launch(void* const* d_in, const int* in_sizes, int n_in,
                              void* d_out, int out_size, void* d_ws, size_t ws_size,
                              hipStream_t stream)
{
    const float* hv    = (const float*)d_in[0];
    const float* he    = (const float*)d_in[1];
    const int*   src   = (const int*)d_in[2];
    const int*   dst   = (const int*)d_in[3];
    const float* W_msg = (const float*)d_in[4];
    const float* b_msg = (const float*)d_in[5];
    const float* W_ih  = (const float*)d_in[6];
    const float* W_hh  = (const float*)d_in[7];
    const float* b_ih  = (const float*)d_in[8];
    const float* b_hh  = (const float*)d_in[9];
    float* out = (float*)d_out;

    // Workspace layout: S1 | S3 | deg | hv_mid
    float* S1     = (float*)d_ws;
    float* S3     = S1  + (size_t)N_NODES * ND;
    float* deg    = S3  + (size_t)N_NODES * ED;
    float* hv_mid = deg + N_NODES;

    const int accum_elems = N_NODES * (ND + ED + 1);   // S1+S3+deg contiguous
    const int zero_blocks = 1024;
    const int edge_blocks = (N_EDGES * 4) / 256;       // 12500, exact
    const int node_blocks = N_NODES / 16;              // 3125, exact

    for (int t = 0; t < ROUNDS; ++t) {
        const float* hin  = (t == 0) ? hv : hv_mid;
        float*       hout = (t == ROUNDS - 1) ? out : hv_mid;

        zero_kernel<<<zero_blocks, 256, 0, stream>>>(S1, accum_elems);

        edge_scatter_kernel<<<edge_blocks, 256, 0, stream>>>(
            hin, he, src, dst, S1, S3, deg, N_EDGES);

        node_update_kernel<<<node_blocks, 256, 0, stream>>>(
            hin, S1, S3, deg,
            W_msg + (size_t)t * CATD * AD,
            b_msg + (size_t)t * AD,
            W_ih  + (size_t)t * AD * H3,
            W_hh  + (size_t)t * ND * H3,
            b_ih  + (size_t)t * H3,
            b_hh  + (size_t)t * H3,
            hout);
    }
}